// SOM_12309376270685
// MI455X (gfx1250) — compile-verified
//
#include <hip/hip_runtime.h>
#include <hip/hip_bf16.h>
#include <math.h>

typedef float v2f __attribute__((ext_vector_type(2)));
typedef float v8f __attribute__((ext_vector_type(8)));

#define THREADS 256
#define SOM_DIM 32
#define LSTRIDE 33   // padded LDS row stride (bank-conflict avoidance)

// ws layout (floats), NB = number of blocks:
//   blkSum  [NB][32]
//   blkMinV [NB]
//   blkMinI [NB] (ints)
//   hdr     [4 + 3*32]: bx, by, radius, sigma2, gbest[32], centroidA[32], centroidB[32]

__global__ __launch_bounds__(THREADS)
void som_pass_a(const float* __restrict__ input,
                const float* __restrict__ particles,
                float* __restrict__ blkSum,
                float* __restrict__ blkMinV,
                int*   __restrict__ blkMinI,
                int N)
{
    __shared__ float stage[THREADS * LSTRIDE];
    __shared__ float wsum[8 * SOM_DIM];
    __shared__ float sdist[THREADS];
    __shared__ int   sidx[THREADS];

    const int t    = threadIdx.x;
    const int p    = blockIdx.x * THREADS + t;
    const int lane = t & 31;
    const int wave = t >> 5;

    float dist2 = 0.f;
    float* srow = &stage[t * LSTRIDE];
    const bool active = (p < N);
    if (active) {
        const float4* x4 = reinterpret_cast<const float4*>(input);
        const float4* p4 = reinterpret_cast<const float4*>(particles + (size_t)p * SOM_DIM);
        #pragma unroll
        for (int q = 0; q < 8; ++q) {
            float4 pv = p4[q];
            float4 xv = x4[q];
            float d0 = pv.x - xv.x, d1 = pv.y - xv.y;
            float d2 = pv.z - xv.z, d3 = pv.w - xv.w;
            dist2 += d0*d0 + d1*d1 + d2*d2 + d3*d3;
            srow[q*4+0] = pv.x; srow[q*4+1] = pv.y;
            srow[q*4+2] = pv.z; srow[q*4+3] = pv.w;
        }
    } else {
        dist2 = 3.402823466e+38f;
        #pragma unroll
        for (int q = 0; q < SOM_DIM; ++q) srow[q] = 0.f;
    }
    __syncthreads();

    // WMMA fp32 ones-reduction: sum this wave's 32 particle rows over all 32 dims.
    // A = ones(16x4) => D[m][n] = sum_k B[k][n]; robust to K-slot placement in B frag.
    const int half = lane >> 4;
    const int n    = lane & 15;
    v2f onesA; onesA.x = 1.0f; onesA.y = 1.0f;
    v8f acc0 = {};
    v8f acc1 = {};
    const int rowbase = wave * 32;
    #pragma unroll
    for (int c = 0; c < 8; ++c) {
        const int r0 = rowbase + c*4 + half;
        const int r1 = r0 + 2;
        v2f b0, b1;
        b0.x = stage[r0*LSTRIDE + n];        b0.y = stage[r1*LSTRIDE + n];
        b1.x = stage[r0*LSTRIDE + 16 + n];   b1.y = stage[r1*LSTRIDE + 16 + n];
        acc0 = __builtin_amdgcn_wmma_f32_16x16x4_f32(false, onesA, false, b0,
                                                     (short)0, acc0, false, false);
        acc1 = __builtin_amdgcn_wmma_f32_16x16x4_f32(false, onesA, false, b1,
                                                     (short)0, acc1, false, false);
    }
    if (lane < 16) {
        // D layout: VGPR r, lanes 0-15 -> M=r, N=lane. Row M=0 carries the column sums.
        wsum[wave*SOM_DIM + lane]      = acc0[0];
        wsum[wave*SOM_DIM + 16 + lane] = acc1[0];
    }

    sdist[t] = dist2;
    sidx[t]  = p;
    __syncthreads();

    // Deterministic tree argmin (tie -> lower index, matching argmin first-occurrence)
    #pragma unroll
    for (int s = 128; s > 0; s >>= 1) {
        if (t < s) {
            float a = sdist[t], b = sdist[t + s];
            int   ia = sidx[t], ib = sidx[t + s];
            if (b < a || (b == a && ib < ia)) { sdist[t] = b; sidx[t] = ib; }
        }
        __syncthreads();
    }

    // Deterministic fixed-order combine of the 8 wave sums
    if (t < SOM_DIM) {
        float s = 0.f;
        #pragma unroll
        for (int w = 0; w < 8; ++w) s += wsum[w*SOM_DIM + t];
        blkSum[(size_t)blockIdx.x * SOM_DIM + t] = s;
    }
    if (t == 0) { blkMinV[blockIdx.x] = sdist[0]; blkMinI[blockIdx.x] = sidx[0]; }
}

__global__ __launch_bounds__(THREADS)
void som_pass_b(const float* __restrict__ particles,
                const int*   __restrict__ gloc,
                const int*   __restrict__ iter_num,
                const float* __restrict__ blkSum,
                const float* __restrict__ blkMinV,
                const int*   __restrict__ blkMinI,
                float* __restrict__ hdr,
                int NB, int N)
{
    __shared__ float sdist[THREADS];
    __shared__ int   sidx[THREADS];
    __shared__ float part[SOM_DIM][8];
    __shared__ int   sbmu;

    const int t = threadIdx.x;

    // global argmin over block partials (deterministic order)
    float best = 3.402823466e+38f; int bidx = 0x7fffffff;
    for (int k = t; k < NB; k += THREADS) {
        float dv = blkMinV[k]; int iv = blkMinI[k];
        if (dv < best || (dv == best && iv < bidx)) { best = dv; bidx = iv; }
    }
    sdist[t] = best; sidx[t] = bidx;
    __syncthreads();
    #pragma unroll
    for (int s = 128; s > 0; s >>= 1) {
        if (t < s) {
            float a = sdist[t], b = sdist[t + s];
            int   ia = sidx[t], ib = sidx[t + s];
            if (b < a || (b == a && ib < ia)) { sdist[t] = b; sidx[t] = ib; }
        }
        __syncthreads();
    }
    if (t == 0) sbmu = sidx[0];

    // deterministic S_all: thread (d = t>>3, seg = t&7) sums a fixed block range
    {
        const int d = t >> 3, seg = t & 7;
        const int per = (NB + 7) >> 3;
        const int b0 = seg * per;
        const int b1 = (b0 + per < NB) ? (b0 + per) : NB;
        float s = 0.f;
        for (int b = b0; b < b1; ++b) s += blkSum[(size_t)b * SOM_DIM + d];
        part[d][seg] = s;
    }
    __syncthreads();

    const int bmu = sbmu;
    if (t < SOM_DIM) {
        float S = 0.f;
        #pragma unroll
        for (int g = 0; g < 8; ++g) S += part[t][g];
        const float gb = particles[(size_t)bmu * SOM_DIM + t];
        hdr[4 + t]              = gb;                          // global_best
        hdr[4 + SOM_DIM + t]    = S / (float)N;                // centroid: all rows
        hdr[4 + 2*SOM_DIM + t]  = (S - gb) / (float)(N - 1);   // centroid: all minus BMU
    }
    if (t == 0) {
        const float decay  = 1.0f - (float)iter_num[0] / 100.0f;  // NUM_ITERS = 100
        const float radius = 0.5f * decay;                        // LEARNING_RADIUS
        const float sig    = 256.0f * decay;                      // SIGMA = max(X,Y)/2
        hdr[0] = (float)gloc[2*bmu];
        hdr[1] = (float)gloc[2*bmu + 1];
        hdr[2] = radius;
        hdr[3] = sig * sig;
    }
}

__global__ __launch_bounds__(THREADS)
void som_pass_c(const float* __restrict__ particles,
                const float* __restrict__ velocities,
                const float* __restrict__ r1,
                const float* __restrict__ r2,
                const int*   __restrict__ gloc,
                const float* __restrict__ hdr,
                float* __restrict__ outP,
                float* __restrict__ outV,
                int N)
{
    const int p = blockIdx.x * THREADS + threadIdx.x;
    if (p >= N) return;

    const float bx = hdr[0], by = hdr[1], radius = hdr[2], sigma2 = hdr[3];
    const float* gbest = hdr + 4;
    const float* cenA  = hdr + 4 + SOM_DIM;
    const float* cenB  = hdr + 4 + 2*SOM_DIM;

    const float4* P4  = (const float4*)(particles  + (size_t)p * SOM_DIM);
    const float4* V4  = (const float4*)(velocities + (size_t)p * SOM_DIM);
    const float4* R14 = (const float4*)(r1 + (size_t)p * SOM_DIM);
    const float4* R24 = (const float4*)(r2 + (size_t)p * SOM_DIM);
    __builtin_prefetch(R14, 0, 0);   // -> global_prefetch_b8
    __builtin_prefetch(R24, 0, 0);

    const int2 gl = *(const int2*)(gloc + 2*(size_t)p);
    const float dx = (float)gl.x - bx;
    const float dy = (float)gl.y - by;
    const float nb = expf(-(dx*dx + dy*dy) / sigma2);
    const bool  skip = (1.0f - nb) > radius;               // global_nbest == 1.0
    const float* cen = ((nb + radius) >= 1.0f) ? cenA : cenB;
    const float4* C4 = (const float4*)cen;
    const float4* G4 = (const float4*)gbest;

    float4* OP = (float4*)(outP + (size_t)p * SOM_DIM);
    float4* OV = (float4*)(outV + (size_t)p * SOM_DIM);

    #pragma unroll
    for (int q = 0; q < 8; ++q) {
        const float4 pv = P4[q], vv = V4[q], a = R14[q], b = R24[q];
        const float4 c = C4[q], g = G4[q];
        float4 vn, pn;
        vn.x = 0.001f*vv.x + 0.01f*a.x*(c.x - pv.x) + 0.1f*b.x*(g.x - pv.x);
        vn.y = 0.001f*vv.y + 0.01f*a.y*(c.y - pv.y) + 0.1f*b.y*(g.y - pv.y);
        vn.z = 0.001f*vv.z + 0.01f*a.z*(c.z - pv.z) + 0.1f*b.z*(g.z - pv.z);
        vn.w = 0.001f*vv.w + 0.01f*a.w*(c.w - pv.w) + 0.1f*b.w*(g.w - pv.w);
        pn.x = pv.x + vn.x; pn.y = pv.y + vn.y;
        pn.z = pv.z + vn.z; pn.w = pv.w + vn.w;
        if (skip) { vn = vv; pn = pv; }
        OP[q] = pn;
        OV[q] = vn;
    }
}

extern "C" void kernel_launch(void* const* d_in, const int* in_sizes, int n_in,
                              void* d_out, int out_size, void* d_ws, size_t ws_size,
                              hipStream_t stream)
{
    (void)n_in; (void)out_size; (void)ws_size;
    const float* input      = (const float*)d_in[0];
    const float* particles  = (const float*)d_in[1];
    const float* velocities = (const float*)d_in[2];
    const float* r1         = (const float*)d_in[3];
    const float* r2         = (const float*)d_in[4];
    const int*   gloc       = (const int*)d_in[5];
    const int*   iter_num   = (const int*)d_in[6];

    const int N  = in_sizes[1] / SOM_DIM;           // 262144
    const int NB = (N + THREADS - 1) / THREADS;     // 1024

    float* ws      = (float*)d_ws;
    float* blkSum  = ws;                             // NB*32 floats
    float* blkMinV = ws + (size_t)NB * SOM_DIM;      // NB floats
    int*   blkMinI = (int*)(blkMinV + NB);           // NB ints
    size_t hoff    = (size_t)NB * SOM_DIM + 2 * (size_t)NB;
    hoff = (hoff + 3) & ~(size_t)3;                  // 16B-align header
    float* hdr     = ws + hoff;

    float* outP = (float*)d_out;
    float* outV = outP + (size_t)N * SOM_DIM;

    som_pass_a<<<NB, THREADS, 0, stream>>>(input, particles, blkSum, blkMinV, blkMinI, N);
    som_pass_b<<<1, THREADS, 0, stream>>>(particles, gloc, iter_num,
                                          blkSum, blkMinV, blkMinI, hdr, NB, N);
    som_pass_c<<<NB, THREADS, 0, stream>>>(particles, velocities, r1, r2,
                                           gloc, hdr, outP, outV, N);
}